// ReLU2D_51651276701788
// MI455X (gfx1250) — compile-verified
//
#include <hip/hip_runtime.h>

// Zonotope ReLU transformer (DeepZ-style), CDNA5 / gfx1250.
//
// Cost model (MI455X): output is 8257*8192 f32 = 270 MB -> ~11.6 us write
// floor at 23.3 TB/s. Everything else (2.1 MB input, 128 KB stats) is noise.
// So: one streaming NT-store pass over the output, fused scatter, and the
// radius abs-reduction mapped onto V_WMMA_F32_16X16X4_F32 (f32-exact path).

typedef __attribute__((ext_vector_type(2))) float v2f;
typedef __attribute__((ext_vector_type(4))) float v4f;
typedef __attribute__((ext_vector_type(8))) float v8f;
typedef __attribute__((ext_vector_type(4))) int   v4i;

#define F_DIM 8192   // CH*NN*NN = 32*16*16
#define G_DIM 65     // generators incl. center
#define OUT_ROWS (G_DIM + F_DIM)

// ---------------------------------------------------------------------------
// Kernel A: radius[f] = sum_{g=1..64} |x[g,f]| via V_WMMA_F32_16X16X4_F32.
// Each wave32 computes one 16-feature tile: A = 16x4 |x| chunk, B = ones,
// 16 K-steps of K=4 accumulate the 64-generator reduction in the f32 C tile.
// A-matrix 16x4 f32 layout: lanes 0-15 hold (M=lane, K={k0,k0+1}) in 2 VGPRs,
// lanes 16-31 hold (M=lane-16, K={k0+2,k0+3}). B is all-ones (layout-free).
// ---------------------------------------------------------------------------
__global__ __launch_bounds__(256) void radius_wmma(const float* __restrict__ x,
                                                   float* __restrict__ radius) {
  const int lane  = threadIdx.x & 31;
  const int wave  = threadIdx.x >> 5;
  const int tile  = blockIdx.x * (blockDim.x >> 5) + wave;  // 512 tiles
  const int f0    = tile << 4;
  const int m     = lane & 15;   // feature row within tile
  const int khalf = lane >> 4;   // which K-pair this half-wave supplies

  v8f acc = {0.f, 0.f, 0.f, 0.f, 0.f, 0.f, 0.f, 0.f};
  const v2f ones = {1.0f, 1.0f};

  #pragma unroll
  for (int kc = 0; kc < 16; ++kc) {
    const int g = 1 + kc * 4 + khalf * 2;   // generators 1..64
    v2f a;
    a.x = __builtin_fabsf(x[(size_t)g       * F_DIM + f0 + m]);
    a.y = __builtin_fabsf(x[(size_t)(g + 1) * F_DIM + f0 + m]);
    // D = A x B + C  (8 args: neg_a, A, neg_b, B, c_mod, C, reuse_a, reuse_b)
    acc = __builtin_amdgcn_wmma_f32_16x16x4_f32(false, a, false, ones,
                                                (short)0, acc, false, false);
  }

  // C/D 16x16 f32 layout: VGPR r = row r (lanes 0-15) / row r+8 (lanes 16-31);
  // every column N is identical (B was ones), so lanes 0 and 16 hold the tile.
  if (m == 0) {
    #pragma unroll
    for (int r = 0; r < 8; ++r)
      radius[f0 + khalf * 8 + r] = acc[r];
  }
}

// ---------------------------------------------------------------------------
// Kernel B: per-feature stats (val/scale/cross) + block-wide prefix sum for
// rank. One 1024-thread workgroup, 8 features/thread, Hillis-Steele scan.
// center[f] = x.reshape(G,F)[0,f] = x[f].
// ---------------------------------------------------------------------------
__global__ __launch_bounds__(1024) void stats_scan(const float* __restrict__ x,
                                                   const float* __restrict__ lambdas,
                                                   const float* __restrict__ radius,
                                                   float* __restrict__ val_out,
                                                   float* __restrict__ scale_out,
                                                   int* __restrict__ rank_out) {
  __shared__ int ssum[1024];
  const int tid  = threadIdx.x;
  const int base = tid * 8;

  float val[8], scl[8];
  int   cross[8];
  int   tsum = 0;

  #pragma unroll
  for (int i = 0; i < 8; ++i) {
    const int   f      = base + i;
    const float c      = x[f];
    const float r      = radius[f];
    const float l      = c - r;
    const float u      = c + r;
    const bool  zero_m = (u <= 0.0f);
    const bool  cr     = (!zero_m) && (l < 0.0f);
    const float denom  = (u - l > 0.0f) ? (u - l) : 1.0f;
    const float slope  = u / denom;
    const float lam    = lambdas[f];
    float v = (lam >= slope) ? (-l * lam * 0.5f) : (u * (1.0f - lam) * 0.5f);
    v        = cr ? v : 0.0f;
    val[i]   = v;
    scl[i]   = zero_m ? 0.0f : (cr ? lam : 1.0f);
    cross[i] = cr ? 1 : 0;
    tsum    += cross[i];
  }

  ssum[tid] = tsum;
  __syncthreads();
  #pragma unroll
  for (int off = 1; off < 1024; off <<= 1) {
    int v = (tid >= off) ? ssum[tid - off] : 0;
    __syncthreads();
    ssum[tid] += v;
    __syncthreads();
  }

  int run = ssum[tid] - tsum;  // exclusive prefix at this thread's first feature
  #pragma unroll
  for (int i = 0; i < 8; ++i) {
    const int f = base + i;
    run += cross[i];
    rank_out[f]  = cross[i] ? (run - 1) : -1;  // -1 never matches an eps row
    val_out[f]   = val[i];
    scale_out[f] = scl[i];
  }
}

// ---------------------------------------------------------------------------
// Kernel C: single streaming pass over the 270 MB output with b128 NT stores.
//   rows [0,65):      out[g,f] = x[g,f]*scale[f] (+ val[f] on row 0)
//   rows [65,65+F):   out[65+c,f] = (rank[f]==c) ? val[f] : 0
// rank/val/scale (96 KB) stay cache-resident; the output stream bypasses
// temporal caching (256 MB > 192 MB L2).
// ---------------------------------------------------------------------------
__global__ __launch_bounds__(256) void write_out(const float* __restrict__ x,
                                                 const float* __restrict__ val,
                                                 const float* __restrict__ scale,
                                                 const int* __restrict__ rank,
                                                 float* __restrict__ out) {
  const size_t total4 = (size_t)OUT_ROWS * (F_DIM / 4);
  const size_t stride = (size_t)gridDim.x * blockDim.x;
  for (size_t idx = (size_t)blockIdx.x * blockDim.x + threadIdx.x;
       idx < total4; idx += stride) {
    const int row = (int)(idx >> 11);           // F/4 = 2048 vec4 per row
    const int f   = ((int)idx & 2047) << 2;
    v4f o;
    if (row < G_DIM) {
      const v4f xv = *(const v4f*)(x + (size_t)row * F_DIM + f);
      const v4f sv = *(const v4f*)(scale + f);
      o = xv * sv;
      if (row == 0) {
        const v4f vv = *(const v4f*)(val + f);
        o += vv;
      }
    } else {
      const int c  = row - G_DIM;
      const v4i rv = *(const v4i*)(rank + f);
      const v4f vv = *(const v4f*)(val + f);
      o.x = (rv.x == c) ? vv.x : 0.0f;
      o.y = (rv.y == c) ? vv.y : 0.0f;
      o.z = (rv.z == c) ? vv.z : 0.0f;
      o.w = (rv.w == c) ? vv.w : 0.0f;
    }
    __builtin_nontemporal_store(o, (v4f*)(out + idx * 4));
  }
}

// ---------------------------------------------------------------------------
extern "C" void kernel_launch(void* const* d_in, const int* in_sizes, int n_in,
                              void* d_out, int out_size, void* d_ws, size_t ws_size,
                              hipStream_t stream) {
  const float* x       = (const float*)d_in[0];   // (65, 32, 16, 16) f32
  const float* lambdas = (const float*)d_in[1];   // (8192,) f32
  float*       out     = (float*)d_out;           // (8257, 32, 16, 16) f32

  float* radius = (float*)d_ws;                   // 8192 f32
  float* val    = radius + F_DIM;                 // 8192 f32
  float* scl    = val + F_DIM;                    // 8192 f32
  int*   rank   = (int*)(scl + F_DIM);            // 8192 i32  (128 KB total)

  // A: 512 feature tiles, 8 waves/block -> 64 blocks.
  radius_wmma<<<64, 256, 0, stream>>>(x, radius);
  // B: one workgroup scan over all 8192 features.
  stats_scan<<<1, 1024, 0, stream>>>(x, lambdas, radius, val, scl, rank);
  // C: bandwidth-bound output pass (~16.9M float4 stores).
  write_out<<<16384, 256, 0, stream>>>(x, val, scl, rank, out);
}